// ATNLPmodel_84078279786829
// MI455X (gfx1250) — compile-verified
//
#include <hip/hip_runtime.h>
#include <stdint.h>

// Problem constants (from reference: B=4, L1=2048, R=4, K=200000, C=32, L2=16)
#define C_DIM   32
#define L2_DIM  16
#define CL      512            // C*L2, flattened inner dim
#define B_DIM   4
#define R_DIM   4
#define BR      16             // B*R rows of the A matrix
#define L1_DIM  2048
#define TILE_K  16             // k-columns per wave tile (WMMA N)
#define WAVES   8
#define THREADS (WAVES * 32)

// LDS staging: chunk = 16 rows x 16 kk f32 = 1KB, padded row stride 20 floats
#define CHUNK_KK   16
#define NCHUNK     (CL / CHUNK_KK)        // 32 chunks per tile
#define B_STRIDE   20                     // padded LDS row stride (floats)
#define CHUNK_LDSF (TILE_K * B_STRIDE)    // 320 floats per chunk buffer

typedef float v2f __attribute__((ext_vector_type(2)));
typedef float v8f __attribute__((ext_vector_type(8)));
typedef int   v4i __attribute__((vector_size(16)));

// pointer types matching the async-to-LDS builtin signature
using gptr_v4i = __attribute__((address_space(1))) v4i*;
using lptr_v4i = __attribute__((address_space(3))) v4i*;

#if __has_builtin(__builtin_amdgcn_global_load_async_to_lds_b128)
#define HAVE_ASYNC 1
#else
#define HAVE_ASYNC 0
#endif

__device__ __forceinline__ void copy16_g2l(const float* g, float* l) {
#if HAVE_ASYNC
  // async DMA: global -> LDS, 16B per lane, tracked by ASYNCcnt
  __builtin_amdgcn_global_load_async_to_lds_b128(
      (gptr_v4i)(uintptr_t)g, (lptr_v4i)(uintptr_t)l, 0, 0);
#else
  *(float4*)l = *(const float4*)g;   // sync fallback (load + ds_store)
#endif
}

template <int N>
__device__ __forceinline__ void wait_async() {
#if HAVE_ASYNC
#if __has_builtin(__builtin_amdgcn_s_wait_asynccnt)
  __builtin_amdgcn_s_wait_asynccnt(N);
#else
  asm volatile("s_wait_asynccnt %0" ::"i"(N));
#endif
  asm volatile("" ::: "memory");     // keep LDS reads below the wait
#endif
}

__global__ __launch_bounds__(THREADS) void atnlp_wmma_kernel(
    const int* __restrict__ seq_input,   // (4, 2048)
    const int* __restrict__ kp_start,    // (4, 4)
    const int* __restrict__ kp_len,      // (4, 4)
    const float* __restrict__ database,  // (K, 32, 16) contiguous
    float* __restrict__ out,             // (4, K)
    int K)
{
  // A matrix stored transposed: lds_At[flat][br], flat = c*16+l  -> conflict-free reads
  __shared__ float lds_At[CL * BR];                       // 32 KB
  __shared__ float lds_B[WAVES][2][CHUNK_LDSF];           // 8*2*320*4 = 20 KB

  const int tid = threadIdx.x;

  // ---- Build one-hot A (16 x 512) from token lookups, transposed into LDS ----
  for (int i = 0; i < (BR * CL) / THREADS; ++i) {         // 32 iters
    int e    = i * THREADS + tid;                         // e = flat*16 + br
    int flat = e >> 4;
    int br   = e & 15;
    int c    = flat >> 4;
    int l    = flat & 15;
    int b    = br >> 2, r = br & 3;
    float start = (float)kp_start[b * R_DIM + r];
    float len   = (float)kp_len[b * R_DIM + r];
    float pos   = start + len * ((float)l / 15.0f);       // matches jnp f32 math
    int idx = (int)floorf(pos);
    idx = idx < 0 ? 0 : (idx > L1_DIM - 1 ? L1_DIM - 1 : idx);
    int tok = seq_input[b * L1_DIM + idx];
    lds_At[e] = (tok == c) ? 1.0f : 0.0f;
  }
  __syncthreads();

  const int wave = tid >> 5;
  const int lane = tid & 31;
  const int n    = lane & 15;        // WMMA N index (k within tile) / A-matrix M row
  const int half = lane >> 4;        // 0: K0/K1 , 1: K2/K3 fragment half
  const int kd0  = half * 2;

  float* buf0 = &lds_B[wave][0][0];
  float* buf1 = &lds_B[wave][1][0];

  const int ntiles = K / TILE_K;     // 12500

  for (int tile = blockIdx.x * WAVES + wave; tile < ntiles;
       tile += gridDim.x * WAVES) {
    const float* gtile = database + (size_t)tile * (TILE_K * CL);  // contiguous 32KB

    // issue one 1KB chunk (16 rows x 16 floats) -> padded LDS buffer; 2 async ops/wave
    auto issue_chunk = [&](int cc, float* buf) {
#pragma unroll
      for (int op = 0; op < 2; ++op) {
        int f   = (op * 32 + lane) * 4;   // float index within chunk [0,256)
        int row = f >> 4;                 // 16 floats per row
        int col = f & 15;
        copy16_g2l(gtile + (size_t)row * CL + cc * CHUNK_KK + col,
                   buf + row * B_STRIDE + col);
      }
    };

    issue_chunk(0, buf0);
    issue_chunk(1, buf1);

    v8f acc = {0.f, 0.f, 0.f, 0.f, 0.f, 0.f, 0.f, 0.f};

    for (int c = 0; c < NCHUNK; ++c) {
      if (c < NCHUNK - 1) wait_async<2>();   // oldest chunk landed (in-order)
      else                wait_async<0>();
      const float* buf = (c & 1) ? buf1 : buf0;
#pragma unroll
      for (int s = 0; s < CHUNK_KK / 4; ++s) {   // 4 WMMA (K=4) per chunk
        const int kkloc = s * 4;
        const int kkabs = c * CHUNK_KK + kkloc;
        v2f a, b;
        // A fragment: A(M=n, K=kd0..kd0+1), transposed storage -> bank-conflict-free
        a.x = lds_At[(kkabs + kd0) * BR + n];
        a.y = lds_At[(kkabs + kd0 + 1) * BR + n];
        // B fragment: B(K=kd0..kd0+1, N=n) = db[kbase+n][kkabs+kd..]
        b.x = buf[n * B_STRIDE + kkloc + kd0];
        b.y = buf[n * B_STRIDE + kkloc + kd0 + 1];
        acc = __builtin_amdgcn_wmma_f32_16x16x4_f32(
            /*neg_a=*/false, a, /*neg_b=*/false, b,
            /*c_mod=*/(short)0, acc, /*reuse_a=*/false, /*reuse_b=*/false);
      }
      if (c + 2 < NCHUNK) issue_chunk(c + 2, (c & 1) ? buf1 : buf0);
    }

    // ---- fold 16 br-rows into 4 b-rows and store ----
    // C/D layout: VGPR v: lanes0-15 -> M=v, lanes16-31 -> M=v+8; M = b*4 + r
    float s0 = acc[0] + acc[1] + acc[2] + acc[3];   // b = half*2
    float s1 = acc[4] + acc[5] + acc[6] + acc[7];   // b = half*2 + 1
    int kcol = tile * TILE_K + n;
    out[(size_t)(half * 2 + 0) * K + kcol] = s0;
    out[(size_t)(half * 2 + 1) * K + kcol] = s1;
  }
}

extern "C" void kernel_launch(void* const* d_in, const int* in_sizes, int n_in,
                              void* d_out, int out_size, void* d_ws, size_t ws_size,
                              hipStream_t stream) {
  const int*   seq = (const int*)d_in[0];
  const int*   kps = (const int*)d_in[1];
  const int*   kpl = (const int*)d_in[2];
  const float* db  = (const float*)d_in[3];
  float*       out = (float*)d_out;

  const int K = in_sizes[3] / (C_DIM * L2_DIM);   // 200000

  dim3 grid(320), block(THREADS);
  hipLaunchKernelGGL(atnlp_wmma_kernel, grid, block, 0, stream,
                     seq, kps, kpl, db, out, K);
}